// MahalanobisDistance_29892972380310
// MI455X (gfx1250) — compile-verified
//
#include <hip/hip_runtime.h>

typedef __attribute__((ext_vector_type(2))) float v2f;
typedef __attribute__((ext_vector_type(8))) float v8f;

#define BATCH 16
#define NPATCH 3136
#define DDIM 256
#define LSTRIDE 17   // pad so deltaT[d][b] rows land on distinct banks

__launch_bounds__(256, 2)
__global__ void mahal_wmma_kernel(const float* __restrict__ features,
                                  const float* __restrict__ mean,
                                  const float* __restrict__ cov_inv,
                                  float* __restrict__ out) {
    __shared__ float deltaT[DDIM * LSTRIDE]; // deltaT[d*17 + b] = features[b][n][d] - mean[n][d]
    __shared__ float dist2[BATCH];

    const int n = blockIdx.x;
    const int t = threadIdx.x;

    if (t < BATCH) dist2[t] = 0.0f;

    // ---- Stage delta (transposed) into LDS: coalesced 1KB rows, conflict-free stores ----
    {
        const float mv = mean[(size_t)n * DDIM + t]; // t == d
        #pragma unroll
        for (int b = 0; b < BATCH; ++b) {
            const float f = features[(size_t)b * NPATCH * DDIM + (size_t)n * DDIM + t];
            deltaT[t * LSTRIDE + b] = f - mv;
        }
    }
    __syncthreads();

    const int wave = t >> 5;
    const int lane = t & 31;
    const int half = lane >> 4;   // 0: lanes 0-15, 1: lanes 16-31
    const int m    = lane & 15;
    const int col0 = wave * 32;   // this wave's 32 output columns (two 16-wide tiles)

    const float* __restrict__ covn = cov_inv + (size_t)n * DDIM * DDIM;

    v8f acc0 = {};
    v8f acc1 = {};

    // left[16 x 256] = delta[16 x 256] x cov_inv[256 x 256], K in steps of 4
    #pragma unroll 8
    for (int k = 0; k < DDIM; k += 4) {
        const int k0 = k + 2 * half; // A/B fragment rows for this half-wave

        // A fragment (16x4 fp32): VGPR v -> delta[m][k0 + v]
        v2f a;
        a.x = deltaT[k0 * LSTRIDE + m];
        a.y = deltaT[(k0 + 1) * LSTRIDE + m];

        // B fragments (4x16 fp32): VGPR v -> cov[k0 + v][col]
        const float* __restrict__ r0 = covn + (size_t)k0 * DDIM;
        v2f b0, b1;
        b0.x = r0[col0 + m];
        b0.y = r0[DDIM + col0 + m];
        b1.x = r0[col0 + 16 + m];
        b1.y = r0[DDIM + col0 + 16 + m];

        acc0 = __builtin_amdgcn_wmma_f32_16x16x4_f32(false, a, false, b0,
                                                     (short)0, acc0, false, false);
        acc1 = __builtin_amdgcn_wmma_f32_16x16x4_f32(false, a, false, b1,
                                                     (short)0, acc1, false, false);
    }

    // ---- dist2[b] += sum_e left[b][e] * delta[b][e] ----
    // C/D layout: VGPR v, lane L -> row (v + 8*half), col (tile_col0 + m)
    #pragma unroll
    for (int v = 0; v < 8; ++v) {
        const int row = v + 8 * half;
        float p = acc0[v] * deltaT[(col0 + m) * LSTRIDE + row]
                + acc1[v] * deltaT[(col0 + 16 + m) * LSTRIDE + row];
        // reduce over the 16 lanes of this half (xor of low 4 lane bits only)
        p += __shfl_xor(p, 1, 32);
        p += __shfl_xor(p, 2, 32);
        p += __shfl_xor(p, 4, 32);
        p += __shfl_xor(p, 8, 32);
        if (m == 0) atomicAdd(&dist2[row], p); // ds_add_f32, rows disjoint within wave
    }
    __syncthreads();

    if (t < BATCH) {
        const float d2 = dist2[t];
        out[(size_t)t * NPATCH + n] = sqrtf(fmaxf(d2, 0.0f));
    }
}

extern "C" void kernel_launch(void* const* d_in, const int* in_sizes, int n_in,
                              void* d_out, int out_size, void* d_ws, size_t ws_size,
                              hipStream_t stream) {
    const float* features = (const float*)d_in[0]; // (16, 3136, 256)
    const float* mean     = (const float*)d_in[1]; // (3136, 256)
    const float* cov_inv  = (const float*)d_in[2]; // (3136, 256, 256)
    float* out            = (float*)d_out;         // (16, 56, 56) == (16, 3136)

    mahal_wmma_kernel<<<NPATCH, 256, 0, stream>>>(features, mean, cov_inv, out);
}